// DecodeDetections_68496138436838
// MI455X (gfx1250) — compile-verified
//
#include <hip/hip_runtime.h>
#include <cstdint>
#include <cmath>

#define BATCH   32
#define NBOX    8732
#define NCLS    20          // foreground classes (1..20)
#define PRED    33
#define NMS_MAX 400
#define TOPK    200
#define CONF_TH 0.01f
#define IOU_TH  0.45f
#define NEG_INF (-__builtin_inff())
#define ELEMS   9           // ceil(8732 / 1024)

// ---------------------------------------------------------------------------
// Kernel 1: decode boxes (B,N,4) + masked transposed scores (B,NCLS,N)
// ---------------------------------------------------------------------------
__global__ void decode_k(const float* __restrict__ y,
                         float* __restrict__ boxes,
                         float* __restrict__ scores) {
  int gid = blockIdx.x * blockDim.x + threadIdx.x;
  if (gid >= BATCH * NBOX) return;
  int b = gid / NBOX;
  int n = gid - b * NBOX;
  const float* p = y + (size_t)gid * PRED;
  __builtin_prefetch(p + PRED * 256, 0, 1);   // gfx1250 global_prefetch_b8

  float l0 = p[21], l1 = p[22], l2 = p[23], l3 = p[24];
  float acx = p[25], acy = p[26], aw = p[27], ah = p[28];
  float v0 = p[29], v1 = p[30], v2 = p[31], v3 = p[32];
  float cx = l0 * v0 * aw + acx;
  float cy = l1 * v1 * ah + acy;
  float w  = expf(l2 * v2) * aw;
  float h  = expf(l3 * v3) * ah;
  float xmin = (cx - 0.5f * w) * 300.0f;
  float ymin = (cy - 0.5f * h) * 300.0f;
  float xmax = (cx + 0.5f * w) * 300.0f;
  float ymax = (cy + 0.5f * h) * 300.0f;
  ((float4*)boxes)[gid] = make_float4(xmin, ymin, xmax, ymax);

#pragma unroll
  for (int c = 0; c < NCLS; ++c) {
    float v = p[1 + c];                       // class 1..20
    scores[((size_t)b * NCLS + c) * NBOX + n] = (v > CONF_TH) ? v : NEG_INF;
  }
}

// payload reduce: (value desc, index asc) + selected box coords, wave32
#define PAYLOAD_REDUCE()                                                    \
  _Pragma("unroll")                                                         \
  for (int m_ = 16; m_ > 0; m_ >>= 1) {                                     \
    float ov = __shfl_xor(bv, m_, 32);                                      \
    int   oi = __shfl_xor(bi, m_, 32);                                      \
    float o0 = __shfl_xor(b0, m_, 32);                                      \
    float o1 = __shfl_xor(b1, m_, 32);                                      \
    float o2 = __shfl_xor(b2, m_, 32);                                      \
    float o3 = __shfl_xor(b3, m_, 32);                                      \
    if (ov > bv || (ov == bv && oi < bi)) {                                 \
      bv = ov; bi = oi; b0 = o0; b1 = o1; b2 = o2; b3 = o3;                 \
    }                                                                       \
  }

// ---------------------------------------------------------------------------
// Kernel 2: greedy NMS, one block (32 waves) per (batch, class) pair.
// Scores staged into LDS via async global->LDS, then everything lives in VGPRs.
// ---------------------------------------------------------------------------
__global__ __launch_bounds__(1024)
void nms_k(const float* __restrict__ boxes,
           const float* __restrict__ scores,
           float* __restrict__ rows) {
  __shared__ float4 s_sc4[NBOX / 4];          // 2183 float4 = 8732 floats
  __shared__ float  s_pv[32];
  __shared__ int    s_pi[32];
  __shared__ float  s_pb[32][4];
  __shared__ float  s_fv;
  __shared__ int    s_fi;
  __shared__ float  s_fb[4];

  const int t    = threadIdx.x;
  const int wave = t >> 5, lane = t & 31;
  const int bc   = blockIdx.x;
  const int b    = bc / NCLS;
  const int c    = bc - b * NCLS;
  const float*  sc_g = scores + (size_t)bc * NBOX;
  const float4* bx_g = (const float4*)boxes + (size_t)b * NBOX;
  float* rows_bc = rows + (size_t)bc * NMS_MAX * 6;

  // --- async stage masked scores into LDS (overlaps with box loads) -------
  {
    unsigned lbase = (unsigned)(uintptr_t)(void*)&s_sc4[0];   // low 32 bits = LDS offset
#pragma unroll
    for (int q = 0; q < 3; ++q) {
      int f = t + (q << 10);
      if (f < NBOX / 4) {
        unsigned laddr = lbase + (unsigned)(f * 16);
        const float4* g = (const float4*)sc_g + f;
        asm volatile("global_load_async_to_lds_b128 %0, %1, off"
                     :: "v"(laddr), "v"(g) : "memory");
      }
    }
  }

  // --- boxes -> registers (overlapped with the async copy) ----------------
  float sc[ELEMS], x1r[ELEMS], y1r[ELEMS], x2r[ELEMS], y2r[ELEMS], ar[ELEMS];
#pragma unroll
  for (int k = 0; k < ELEMS; ++k) {
    int i = t + (k << 10);
    float4 bb = (i < NBOX) ? bx_g[i] : make_float4(0.f, 0.f, 0.f, 0.f);
    x1r[k] = bb.x; y1r[k] = bb.y; x2r[k] = bb.z; y2r[k] = bb.w;
    ar[k]  = fmaxf(bb.z - bb.x, 0.f) * fmaxf(bb.w - bb.y, 0.f);
  }

#if __has_builtin(__builtin_amdgcn_s_wait_asynccnt)
  __builtin_amdgcn_s_wait_asynccnt(0);
#else
  asm volatile("s_wait_asynccnt 0" ::: "memory");
#endif
  __syncthreads();

  const float* s_sc = (const float*)s_sc4;
#pragma unroll
  for (int k = 0; k < ELEMS; ++k) {
    int i = t + (k << 10);
    sc[k] = (i < NBOX) ? s_sc[i] : NEG_INF;
  }

  const float cls_f = (float)(c + 1);
  int s = 0;
  for (; s < NMS_MAX; ++s) {
    // local argmax over this thread's 9 elements (lowest-index ties)
    float bv = sc[0]; int bi = t;
    float b0 = x1r[0], b1 = y1r[0], b2 = x2r[0], b3 = y2r[0];
#pragma unroll
    for (int k = 1; k < ELEMS; ++k) {
      if (sc[k] > bv) {
        bv = sc[k]; bi = t + (k << 10);
        b0 = x1r[k]; b1 = y1r[k]; b2 = x2r[k]; b3 = y2r[k];
      }
    }
    PAYLOAD_REDUCE();
    if (lane == 0) {
      s_pv[wave] = bv; s_pi[wave] = bi;
      s_pb[wave][0] = b0; s_pb[wave][1] = b1; s_pb[wave][2] = b2; s_pb[wave][3] = b3;
    }
    __syncthreads();
    if (wave == 0) {
      bv = s_pv[lane]; bi = s_pi[lane];
      b0 = s_pb[lane][0]; b1 = s_pb[lane][1]; b2 = s_pb[lane][2]; b3 = s_pb[lane][3];
      PAYLOAD_REDUCE();
      if (lane == 0) {
        s_fv = bv; s_fi = bi;
        s_fb[0] = b0; s_fb[1] = b1; s_fb[2] = b2; s_fb[3] = b3;
      }
    }
    __syncthreads();

    float selv = s_fv;
    int   seli = s_fi;
    float sx1 = s_fb[0], sy1 = s_fb[1], sx2 = s_fb[2], sy2 = s_fb[3];

    if (!(selv > NEG_INF)) break;             // uniform: nothing valid remains

    if (t == 0) {
      float* r = rows_bc + s * 6;
      r[0] = cls_f; r[1] = selv; r[2] = sx1; r[3] = sy1; r[4] = sx2; r[5] = sy2;
    }

    float sarea = fmaxf(sx2 - sx1, 0.f) * fmaxf(sy2 - sy1, 0.f);
#pragma unroll
    for (int k = 0; k < ELEMS; ++k) {
      int i = t + (k << 10);
      if (i == seli) {
        sc[k] = NEG_INF;                      // s.at[idx].set(-inf)
      } else if (sc[k] > NEG_INF) {
        float ix1 = fmaxf(sx1, x1r[k]);
        float iy1 = fmaxf(sy1, y1r[k]);
        float ix2 = fminf(sx2, x2r[k]);
        float iy2 = fminf(sy2, y2r[k]);
        float inter = fmaxf(ix2 - ix1, 0.f) * fmaxf(iy2 - iy1, 0.f);
        float uni   = sarea + ar[k] - inter;
        float iou   = (uni > 0.f) ? inter / uni : 0.f;
        if (iou > IOU_TH) sc[k] = NEG_INF;
      }
    }
  }
  // zero-fill rows for remaining (invalid) steps
  for (int e = s * 6 + t; e < NMS_MAX * 6; e += 1024) rows_bc[e] = 0.f;
}

// ---------------------------------------------------------------------------
// Kernel 3: per-batch top-200 by confidence (iterative argmax, ties -> low idx)
// ---------------------------------------------------------------------------
__global__ __launch_bounds__(1024)
void topk_k(const float* __restrict__ rows, float* __restrict__ out) {
  __shared__ float s_conf[NCLS * NMS_MAX];    // 8000 floats
  __shared__ float s_pv[32];
  __shared__ int   s_pi[32];
  __shared__ int   s_sel[TOPK];

  const int t = threadIdx.x, b = blockIdx.x;
  const int wave = t >> 5, lane = t & 31;
  const float* rb = rows + (size_t)b * (NCLS * NMS_MAX) * 6;

  for (int r = t; r < NCLS * NMS_MAX; r += 1024) s_conf[r] = rb[r * 6 + 1];
  __syncthreads();

  for (int j = 0; j < TOPK; ++j) {
    float bv = s_conf[t]; int bi = t;
    for (int r = t + 1024; r < NCLS * NMS_MAX; r += 1024) {
      float v = s_conf[r];
      if (v > bv) { bv = v; bi = r; }
    }
#pragma unroll
    for (int m = 16; m > 0; m >>= 1) {
      float ov = __shfl_xor(bv, m, 32);
      int   oi = __shfl_xor(bi, m, 32);
      if (ov > bv || (ov == bv && oi < bi)) { bv = ov; bi = oi; }
    }
    if (lane == 0) { s_pv[wave] = bv; s_pi[wave] = bi; }
    __syncthreads();
    if (wave == 0) {
      bv = s_pv[lane]; bi = s_pi[lane];
#pragma unroll
      for (int m = 16; m > 0; m >>= 1) {
        float ov = __shfl_xor(bv, m, 32);
        int   oi = __shfl_xor(bi, m, 32);
        if (ov > bv || (ov == bv && oi < bi)) { bv = ov; bi = oi; }
      }
      if (lane == 0) { s_sel[j] = bi; s_conf[bi] = NEG_INF; }
    }
    __syncthreads();
  }

  float* ob = out + (size_t)b * TOPK * 6;
  for (int e = t; e < TOPK * 6; e += 1024) {
    int j = e / 6, m = e - j * 6;
    ob[e] = rb[s_sel[j] * 6 + m];
  }
}

// ---------------------------------------------------------------------------
extern "C" void kernel_launch(void* const* d_in, const int* in_sizes, int n_in,
                              void* d_out, int out_size, void* d_ws, size_t ws_size,
                              hipStream_t stream) {
  (void)in_sizes; (void)n_in; (void)out_size; (void)ws_size;
  const float* y = (const float*)d_in[0];
  float* boxes  = (float*)d_ws;                               // B*N*4
  float* scores = boxes  + (size_t)BATCH * NBOX * 4;          // B*NCLS*N
  float* rows   = scores + (size_t)BATCH * NCLS * NBOX;       // B*NCLS*NMS_MAX*6
  float* out    = (float*)d_out;

  int total = BATCH * NBOX;
  decode_k<<<(total + 255) / 256, 256, 0, stream>>>(y, boxes, scores);
  nms_k<<<BATCH * NCLS, 1024, 0, stream>>>(boxes, scores, rows);
  topk_k<<<BATCH, 1024, 0, stream>>>(rows, out);
}